// VoxelDownsampling_23562190586069
// MI455X (gfx1250) — compile-verified
//
#include <hip/hip_runtime.h>
#include <hip/hip_bf16.h>
#include <stdint.h>

// ---------------------------------------------------------------------------
// Voxel downsampling for MI455X (gfx1250).
// Memory-bound: ~2x 48MB point reads + ~19MB L2-resident voxel grid atomics
// + 64MB output writes  =>  ~9us floor at 23.3 TB/s. No matmul structure, so
// no WMMA; the CDNA5-specific path is double-buffered async global->LDS DMA
// (GLOBAL_LOAD_ASYNC_TO_LDS_B32 + S_WAIT_ASYNCCNT thresholds) overlapping
// HBM streaming with voxel-key compute + L2 atomics, plus GLOBAL_PREFETCH_B8
// in the min/max pass.
// ---------------------------------------------------------------------------

#define I32_MAX  2147483647
#define I32_MIN  (-2147483647 - 1)

// Workspace layout (bytes):
//   [0,256)      : int header
//   [256,4352)   : 1024 block sums for the compaction scan
//   [16384, ...) : dense voxel grid, float4 per cell {sumx,sumy,sumz,count}
#define HDR_MIN0   0
#define HDR_MAX0   3
#define HDR_K      6
#define HDR_S0     7   // d1*d2
#define HDR_S1     8   // d2
#define HDR_M      9   // total occupied voxels (diagnostic)
#define BSUM_OFF   256
#define GRID_OFF   16384
#define SCAN_BLOCKS 1024
#define TILE        256

// ---------------- gfx1250 feature probes ----------------
#if defined(__has_builtin)
#  if __has_builtin(__builtin_amdgcn_global_load_async_to_lds_b32)
#    define HAS_ASYNC_LDS 1
#  endif
#  if __has_builtin(__builtin_amdgcn_s_wait_asynccnt)
#    define HAS_WAIT_ASYNC 1
#  endif
#endif
#ifndef HAS_ASYNC_LDS
#  define HAS_ASYNC_LDS 0
#endif
#ifndef HAS_WAIT_ASYNC
#  define HAS_WAIT_ASYNC 0
#endif

__device__ __forceinline__ void wait_async_le0() {
#if HAS_WAIT_ASYNC
  __builtin_amdgcn_s_wait_asynccnt(0);
#else
  asm volatile("s_wait_asynccnt 0" ::: "memory");
#endif
}

__device__ __forceinline__ void wait_async_le3() {
#if HAS_WAIT_ASYNC
  __builtin_amdgcn_s_wait_asynccnt(3);
#else
  asm volatile("s_wait_asynccnt 3" ::: "memory");
#endif
}

#if HAS_ASYNC_LDS
// Per-lane 4-byte async copy global -> LDS. Builtin params are
// addrspace(1) int* / addrspace(3) int*; generic LDS pointer low 32 bits are
// the wave-relative LDS byte offset (ISA 10.2), so the inttoptr casts are
// exact. The immediate INST_OFFSET is added to BOTH the global and the LDS
// address (ISA 15.18 async pseudocode), so one address pair serves a whole
// tile with offsets 0/1024/2048.
template <int OFF>
__device__ __forceinline__ void async_g2l_b32(const float* g, const float* l) {
  __builtin_amdgcn_global_load_async_to_lds_b32(
      (__attribute__((address_space(1))) int*)(uintptr_t)g,
      (__attribute__((address_space(3))) int*)(uint32_t)(uintptr_t)l,
      OFF, /*cpol=*/0);
}
#endif

// ---------------- main streaming kernel (kept first for disasm) ----------------
// Double-buffered async-DMA pipeline: prefetch tile t+1 into LDS buffer b^1
// while computing keys + issuing grid atomics for tile t from buffer b.
// Each wave issues exactly 3 async B32 ops per full tile, and async loads
// retire in order, so s_wait_asynccnt<=3 waits exactly for tile t's copies.
__global__ void __launch_bounds__(TILE)
k_accum(const float* __restrict__ pts, const float* __restrict__ leaf_p,
        int n, const int* __restrict__ hdr, float* __restrict__ grid) {
  __shared__ float tile[2][TILE * 3];
  const float leaf = leaf_p[0];
  const int mn0 = hdr[HDR_MIN0 + 0];
  const int mn1 = hdr[HDR_MIN0 + 1];
  const int mn2 = hdr[HDR_MIN0 + 2];
  const int s0 = hdr[HDR_S0];
  const int s1 = hdr[HDR_S1];
  const long long K = hdr[HDR_K];
  const int tid = threadIdx.x;
  const long long step = (long long)gridDim.x * TILE;

  long long tb = (long long)blockIdx.x * TILE;
  int buf = 0;

#if HAS_ASYNC_LDS
  // Preload first full tile.
  if (tb + TILE <= (long long)n) {
    const float* g = &pts[tb * 3 + tid];
    const float* l = &tile[buf][tid];
    async_g2l_b32<0>(g, l);
    async_g2l_b32<1024>(g, l);
    async_g2l_b32<2048>(g, l);
  }
  while (tb + TILE <= (long long)n) {
    long long nx = tb + step;
    bool pre = (nx + TILE <= (long long)n);
    if (pre) {  // issue DMA for the next full tile into the other buffer
      const float* g = &pts[nx * 3 + tid];
      const float* l = &tile[buf ^ 1][tid];
      async_g2l_b32<0>(g, l);
      async_g2l_b32<1024>(g, l);
      async_g2l_b32<2048>(g, l);
      wait_async_le3();   // current tile's 3 older copies have landed
    } else {
      wait_async_le0();
    }
    __syncthreads();

    {
      float x = tile[buf][tid * 3 + 0];
      float y = tile[buf][tid * 3 + 1];
      float z = tile[buf][tid * 3 + 2];
      int cx = (int)floorf(x / leaf) - mn0;
      int cy = (int)floorf(y / leaf) - mn1;
      int cz = (int)floorf(z / leaf) - mn2;
      long long key = (long long)cx * s0 + (long long)cy * s1 + (long long)cz;
      if (key >= 0 && key < K) {
        long long g = key * 4;
        atomicAdd(&grid[g + 0], x);
        atomicAdd(&grid[g + 1], y);
        atomicAdd(&grid[g + 2], z);
        atomicAdd(&grid[g + 3], 1.0f);
      }
    }
    __syncthreads();   // all waves done reading buf before it is re-filled
    buf ^= 1;
    tb = nx;
  }
  // Partial tail tile (block-uniform branch; barriers stay uniform).
  if (tb < (long long)n) {
    int cnt = (int)((long long)n - tb);
    int cnt3 = cnt * 3;
    for (int j = tid; j < cnt3; j += TILE)
      async_g2l_b32<0>(&pts[tb * 3 + j], &tile[buf][j]);
    wait_async_le0();
    __syncthreads();
    if (tid < cnt) {
      float x = tile[buf][tid * 3 + 0];
      float y = tile[buf][tid * 3 + 1];
      float z = tile[buf][tid * 3 + 2];
      int cx = (int)floorf(x / leaf) - mn0;
      int cy = (int)floorf(y / leaf) - mn1;
      int cz = (int)floorf(z / leaf) - mn2;
      long long key = (long long)cx * s0 + (long long)cy * s1 + (long long)cz;
      if (key >= 0 && key < K) {
        long long g = key * 4;
        atomicAdd(&grid[g + 0], x);
        atomicAdd(&grid[g + 1], y);
        atomicAdd(&grid[g + 2], z);
        atomicAdd(&grid[g + 3], 1.0f);
      }
    }
  }
#else
  for (; tb < (long long)n; tb += step) {
    int cnt = ((long long)n - tb < TILE) ? (int)((long long)n - tb) : TILE;
    int cnt3 = cnt * 3;
    for (int j = tid; j < cnt3; j += TILE)
      tile[buf][j] = pts[tb * 3 + j];
    __syncthreads();
    if (tid < cnt) {
      float x = tile[buf][tid * 3 + 0];
      float y = tile[buf][tid * 3 + 1];
      float z = tile[buf][tid * 3 + 2];
      int cx = (int)floorf(x / leaf) - mn0;
      int cy = (int)floorf(y / leaf) - mn1;
      int cz = (int)floorf(z / leaf) - mn2;
      long long key = (long long)cx * s0 + (long long)cy * s1 + (long long)cz;
      if (key >= 0 && key < K) {
        long long g = key * 4;
        atomicAdd(&grid[g + 0], x);
        atomicAdd(&grid[g + 1], y);
        atomicAdd(&grid[g + 2], z);
        atomicAdd(&grid[g + 3], 1.0f);
      }
    }
    __syncthreads();
  }
#endif
}

// ---------------- helper kernels ----------------

__global__ void k_zero_out(float* __restrict__ out, long long n) {
  long long stride = (long long)gridDim.x * blockDim.x;
  for (long long i = (long long)blockIdx.x * blockDim.x + threadIdx.x; i < n; i += stride)
    out[i] = 0.0f;
}

__global__ void k_init(int* __restrict__ hdr) {
  int t = threadIdx.x;
  if (t < 3) {
    hdr[HDR_MIN0 + t] = I32_MAX;
    hdr[HDR_MAX0 + t] = I32_MIN;
  } else if (t < 16) {
    hdr[t] = 0;
  }
}

__global__ void k_minmax(const float* __restrict__ pts,
                         const float* __restrict__ leaf_p,
                         int n, int* __restrict__ hdr) {
  __shared__ int smn[3], smx[3];
  if (threadIdx.x < 3) { smn[threadIdx.x] = I32_MAX; smx[threadIdx.x] = I32_MIN; }
  __syncthreads();
  const float leaf = leaf_p[0];
  int lmn0 = I32_MAX, lmn1 = I32_MAX, lmn2 = I32_MAX;
  int lmx0 = I32_MIN, lmx1 = I32_MIN, lmx2 = I32_MIN;
  long long stride = (long long)gridDim.x * blockDim.x;
  for (long long i = (long long)blockIdx.x * blockDim.x + threadIdx.x; i < n; i += stride) {
    __builtin_prefetch(pts + (i + stride) * 3, 0, 1);  // -> global_prefetch_b8
    float x = pts[i * 3 + 0];
    float y = pts[i * 3 + 1];
    float z = pts[i * 3 + 2];
    int cx = (int)floorf(x / leaf);
    int cy = (int)floorf(y / leaf);
    int cz = (int)floorf(z / leaf);
    lmn0 = min(lmn0, cx); lmx0 = max(lmx0, cx);
    lmn1 = min(lmn1, cy); lmx1 = max(lmx1, cy);
    lmn2 = min(lmn2, cz); lmx2 = max(lmx2, cz);
  }
  atomicMin(&smn[0], lmn0); atomicMax(&smx[0], lmx0);
  atomicMin(&smn[1], lmn1); atomicMax(&smx[1], lmx1);
  atomicMin(&smn[2], lmn2); atomicMax(&smx[2], lmx2);
  __syncthreads();
  if (threadIdx.x < 3) {
    atomicMin(&hdr[HDR_MIN0 + threadIdx.x], smn[threadIdx.x]);
    atomicMax(&hdr[HDR_MAX0 + threadIdx.x], smx[threadIdx.x]);
  }
}

__global__ void k_dims(int* __restrict__ hdr, long long capCells) {
  if (blockIdx.x == 0 && threadIdx.x == 0) {
    int d0 = hdr[HDR_MAX0 + 0] - hdr[HDR_MIN0 + 0] + 1;
    int d1 = hdr[HDR_MAX0 + 1] - hdr[HDR_MIN0 + 1] + 1;
    int d2 = hdr[HDR_MAX0 + 2] - hdr[HDR_MIN0 + 2] + 1;
    if (d0 < 1) d0 = 1;
    if (d1 < 1) d1 = 1;
    if (d2 < 1) d2 = 1;
    long long K = (long long)d0 * (long long)d1 * (long long)d2;
    if (K > capCells) K = capCells;        // safety clamp to workspace size
    if (K > 0x7fffffffLL) K = 0x7fffffffLL;
    if (K < 1) K = 1;
    hdr[HDR_K]  = (int)K;
    hdr[HDR_S0] = d1 * d2;
    hdr[HDR_S1] = d2;
  }
}

__global__ void k_zero_grid(float* __restrict__ grid, const int* __restrict__ hdr) {
  long long tot = 4LL * (long long)hdr[HDR_K];
  long long stride = (long long)gridDim.x * blockDim.x;
  for (long long i = (long long)blockIdx.x * blockDim.x + threadIdx.x; i < tot; i += stride)
    grid[i] = 0.0f;
}

__global__ void k_count(const float* __restrict__ grid,
                        const int* __restrict__ hdr,
                        int* __restrict__ bsum) {
  __shared__ int s[256];
  int K = hdr[HDR_K];
  int chunk = (K + (int)gridDim.x - 1) / (int)gridDim.x;
  int start = chunk * blockIdx.x;
  int end = min(K, start + chunk);
  int c = 0;
  for (int i = start + threadIdx.x; i < end; i += blockDim.x)
    c += (grid[(long long)i * 4 + 3] > 0.0f) ? 1 : 0;
  s[threadIdx.x] = c;
  __syncthreads();
  for (int off = 128; off > 0; off >>= 1) {
    if ((int)threadIdx.x < off) s[threadIdx.x] += s[threadIdx.x + off];
    __syncthreads();
  }
  if (threadIdx.x == 0) bsum[blockIdx.x] = s[0];
}

__global__ void __launch_bounds__(SCAN_BLOCKS)
k_scan(int* __restrict__ bsum, int* __restrict__ hdr) {
  __shared__ int s[SCAN_BLOCKS];
  int t = threadIdx.x;
  int v = bsum[t];
  s[t] = v;
  __syncthreads();
  for (int off = 1; off < SCAN_BLOCKS; off <<= 1) {
    int u = (t >= off) ? s[t - off] : 0;
    __syncthreads();
    s[t] += u;
    __syncthreads();
  }
  bsum[t] = s[t] - v;                 // exclusive scan
  if (t == SCAN_BLOCKS - 1) hdr[HDR_M] = s[t];
}

// One wave32 per chunk: ordered compaction via ballot-prefix, so output rank
// == dense group id in ascending key order (matches the reference argsort).
__global__ void __launch_bounds__(32)
k_emit(const float* __restrict__ grid, const int* __restrict__ hdr,
       const int* __restrict__ bsum, float* __restrict__ out, int n) {
  int K = hdr[HDR_K];
  int chunk = (K + (int)gridDim.x - 1) / (int)gridDim.x;
  int start = chunk * blockIdx.x;
  int end = min(K, start + chunk);
  int base = bsum[blockIdx.x];
  int lane = threadIdx.x;
  unsigned long long lt_mask = (1ull << lane) - 1ull;

  for (int i = start; i < end; i += 32) {
    int idx = i + lane;
    float c = 0.0f;
    bool occ = false;
    if (idx < end) {
      c = grid[(long long)idx * 4 + 3];
      occ = (c > 0.0f);
    }
    unsigned long long m = __ballot(occ);
    if (occ) {
      int r = base + __popcll(m & lt_mask);
      float inv = 1.0f / c;
      long long g = (long long)idx * 4;
      out[(long long)r * 3 + 0] = grid[g + 0] * inv;
      out[(long long)r * 3 + 1] = grid[g + 1] * inv;
      out[(long long)r * 3 + 2] = grid[g + 2] * inv;
      out[(long long)n * 3 + r] = 1.0f;   // mask
    }
    base += __popcll(m);
  }
}

// ---------------- launch ----------------

extern "C" void kernel_launch(void* const* d_in, const int* in_sizes, int n_in,
                              void* d_out, int out_size, void* d_ws, size_t ws_size,
                              hipStream_t stream) {
  const float* pts  = (const float*)d_in[0];
  const float* leaf = (const float*)d_in[1];
  int n = in_sizes[0] / 3;

  int*   hdr  = (int*)d_ws;
  int*   bsum = (int*)((char*)d_ws + BSUM_OFF);
  float* grid = (float*)((char*)d_ws + GRID_OFF);
  long long capCells = ((long long)ws_size - GRID_OFF) / 16;
  if (capCells < 1) capCells = 1;

  k_zero_out <<<2048, 256, 0, stream>>>((float*)d_out, (long long)out_size);
  k_init     <<<1, 32, 0, stream>>>(hdr);
  k_minmax   <<<1024, 256, 0, stream>>>(pts, leaf, n, hdr);
  k_dims     <<<1, 32, 0, stream>>>(hdr, capCells);
  k_zero_grid<<<2048, 256, 0, stream>>>(grid, hdr);
  k_accum    <<<2048, TILE, 0, stream>>>(pts, leaf, n, hdr, grid);
  k_count    <<<SCAN_BLOCKS, 256, 0, stream>>>(grid, hdr, bsum);
  k_scan     <<<1, SCAN_BLOCKS, 0, stream>>>(bsum, hdr);
  k_emit     <<<SCAN_BLOCKS, 32, 0, stream>>>(grid, hdr, bsum, (float*)d_out, n);
}